// SURVEY_BA_2grid_MinCutPool_21930103013660
// MI455X (gfx1250) — compile-verified
//
#include <hip/hip_runtime.h>
#include <hip/hip_bf16.h>
#include <math.h>

// ---------------------------------------------------------------------------
// MinCutPool GNN forward for MI455X (gfx1250, wave32).
// Sparse-adjacency formulation (never materializes [N,N]); dense per-node
// matmuls via V_WMMA_F32_16X16X4_F32 with LDS-staged, zero-padded operand
// tiles (branch-free ds_load_b64 in the K-loop) and wave-uniform full/partial
// tile epilogues (no per-element exec-mask churn on the stores).
// ---------------------------------------------------------------------------

typedef __attribute__((ext_vector_type(2))) float v2f;
typedef __attribute__((ext_vector_type(8))) float v8f;

#define F_IN   10
#define H_DIM  32
#define K1     35   // pool-1 clusters
#define K2     18   // pool-2 clusters

// ---------------------------------------------------------------------------
// 0) init: deg = 1 (self loop), zero the atomic accumulators
// ---------------------------------------------------------------------------
__global__ void k_init(float* __restrict__ deg, float* __restrict__ a_s,
                       float* __restrict__ out1, float* __restrict__ adj1, int N) {
    int idx = blockIdx.x * 256 + threadIdx.x;
    if (idx < N) { deg[idx] = 1.0f; return; }
    idx -= N;
    if (idx < N * K1) { a_s[idx] = 0.0f; return; }
    idx -= N * K1;
    if (idx < K1 * H_DIM) { out1[idx] = 0.0f; return; }
    idx -= K1 * H_DIM;
    if (idx < K1 * K1) { adj1[idx] = 0.0f; }
}

// ---------------------------------------------------------------------------
// 1) degree of targets over E edges; dis = rsqrt(deg)
// ---------------------------------------------------------------------------
__global__ void k_deg(const int* __restrict__ e1, float* __restrict__ deg, int E) {
    int e = blockIdx.x * 256 + threadIdx.x;
    if (e < E) atomicAdd(&deg[e1[e]], 1.0f);
}

__global__ void k_dis(const float* __restrict__ deg, float* __restrict__ dis, int N) {
    int i = blockIdx.x * 256 + threadIdx.x;
    if (i < N) dis[i] = rsqrtf(deg[i]);   // deg >= 1 always (self loops)
}

// ---------------------------------------------------------------------------
// 2) h = x @ W1 (WMMA f32 16x16x4, K 10->12 zero-padded in LDS), and
//    agg init = dis[i]^2 * h[i]  (self-loop term). 4 waves/block, one
//    16-node tile per wave; W1 panel transposed+padded in LDS, shared.
// ---------------------------------------------------------------------------
__global__ void k_xw1(const float* __restrict__ x, const float* __restrict__ W1,
                      const float* __restrict__ dis, float* __restrict__ h,
                      float* __restrict__ agg, int N) {
    __shared__ float w1t[H_DIM * 12];          // W1^T, [col][k], k padded to 12
    __shared__ float xt[4][16 * 12];           // per-wave x tile, stride 12 (even)
    __shared__ float d2[4][16];                // per-wave dis^2 for the 16 nodes
    int tid = threadIdx.x;
    int lane = tid & 31, w = tid >> 5;
    int hl = lane & 15, hi = lane >> 4;
    int n0 = (blockIdx.x * 4 + w) * 16;

    for (int i = tid; i < H_DIM * 12; i += 128) {        // fill W1^T (padded)
        int col = i / 12, k = i % 12;
        w1t[i] = (k < F_IN) ? W1[k * H_DIM + col] : 0.0f;
    }
    for (int i = lane; i < 16 * 12; i += 32) {           // fill x tile (padded)
        int r = i / 12, k = i % 12;
        int nd = n0 + r;
        xt[w][i] = (k < F_IN && nd < N) ? x[nd * F_IN + k] : 0.0f;
    }
    if (lane < 16) {                                     // stage dis^2 once
        int nd = n0 + lane;
        float di = (nd < N) ? dis[nd] : 0.0f;
        d2[w][lane] = di * di;
    }
    __syncthreads();

    bool full = (n0 + 16 <= N);                // wave-uniform -> scalar branch
    for (int ct = 0; ct < 2; ct++) {           // 32 output cols -> 2 tiles of 16
        int col0 = ct * 16;
        v8f acc = {};
        #pragma unroll
        for (int tk = 0; tk < 3; tk++) {       // K = 12 -> 3 steps of 4
            int kb = tk * 4 + hi * 2;          // even -> aligned v2f (ds_load_b64)
            v2f a = *(const v2f*)&xt[w][hl * 12 + kb];
            v2f b = *(const v2f*)&w1t[(col0 + hl) * 12 + kb];
            acc = __builtin_amdgcn_wmma_f32_16x16x4_f32(
                      false, a, false, b, (short)0, acc, false, false);
        }
        int col = col0 + hl;
        if (full) {                            // unguarded fast path
            #pragma unroll
            for (int r = 0; r < 8; r++) {      // C: VGPR r -> M=r / M=r+8
                int m = r + hi * 8;
                int nd = n0 + m;
                float v = acc[r];
                h[nd * H_DIM + col]   = v;
                agg[nd * H_DIM + col] = d2[w][m] * v;
            }
        } else if (n0 < N) {                   // last partial tile only
            for (int r = 0; r < 8; r++) {
                int m = r + hi * 8;
                int nd = n0 + m;
                if (nd < N) {
                    float v = acc[r];
                    h[nd * H_DIM + col]   = v;
                    agg[nd * H_DIM + col] = d2[w][m] * v;
                }
            }
        }
    }
}

// ---------------------------------------------------------------------------
// 3) edge aggregation: agg[c,f] += dis[r]*dis[c] * h[r,f]
// ---------------------------------------------------------------------------
__global__ void k_edge(const int* __restrict__ e0, const int* __restrict__ e1,
                       const float* __restrict__ dis, const float* __restrict__ h,
                       float* __restrict__ agg, int E) {
    unsigned idx = blockIdx.x * 256u + threadIdx.x;
    int e = (int)(idx >> 5);
    int f = (int)(idx & 31u);
    if (e >= E) return;
    int r = e0[e], c = e1[e];
    float norm = dis[r] * dis[c];
    atomicAdd(&agg[c * H_DIM + f], norm * h[r * H_DIM + f]);
}

// ---------------------------------------------------------------------------
// 4) h1 = relu(agg + b1); logits = h1 @ pW1 + pb1 (WMMA, cols 35->48 padded);
//    row-softmax -> s. 4 waves/block; pW1^T panel in LDS shared by the block.
// ---------------------------------------------------------------------------
__global__ void k_h1s(const float* __restrict__ agg, const float* __restrict__ b1,
                      const float* __restrict__ pW1, const float* __restrict__ pb1,
                      float* __restrict__ h1, float* __restrict__ s, int N) {
    __shared__ float pwt[48 * 34];             // pW1^T, [col][k], stride 34 (even)
    __shared__ float tile[4][16 * 34];         // per-wave h1 tile, stride 34
    __shared__ float slog[4][16 * 49];         // per-wave logits, stride 49
    int tid = threadIdx.x;
    int lane = tid & 31, w = tid >> 5;
    int hl = lane & 15, hi = lane >> 4;
    int n0 = (blockIdx.x * 4 + w) * 16;

    for (int i = tid; i < 48 * H_DIM; i += 128) {        // fill pW1^T (padded)
        int col = i >> 5, k = i & 31;
        pwt[col * 34 + k] = (col < K1) ? pW1[k * K1 + col] : 0.0f;
    }
    float bias = b1[lane];
    bool full = (n0 + 16 <= N);                // wave-uniform
    if (full) {
        for (int r = 0; r < 16; r++) {         // relu(agg+b1) tile, fast path
            int nd = n0 + r;
            float v = fmaxf(agg[nd * H_DIM + lane] + bias, 0.0f);
            tile[w][r * 34 + lane] = v;
            h1[nd * H_DIM + lane] = v;
        }
    } else {
        for (int r = 0; r < 16; r++) {
            int nd = n0 + r;
            float v = (nd < N) ? fmaxf(agg[nd * H_DIM + lane] + bias, 0.0f) : 0.0f;
            tile[w][r * 34 + lane] = v;
            if (nd < N) h1[nd * H_DIM + lane] = v;
        }
    }
    __syncthreads();

    for (int ct = 0; ct < 3; ct++) {
        int col0 = ct * 16;
        v8f acc = {};
        #pragma unroll
        for (int tk = 0; tk < 8; tk++) {       // K = 32 -> 8 steps of 4
            int kb = tk * 4 + hi * 2;          // even -> aligned v2f loads
            v2f a = *(const v2f*)&tile[w][hl * 34 + kb];
            v2f b = *(const v2f*)&pwt[(col0 + hl) * 34 + kb];
            acc = __builtin_amdgcn_wmma_f32_16x16x4_f32(
                      false, a, false, b, (short)0, acc, false, false);
        }
        #pragma unroll
        for (int r = 0; r < 8; r++)
            slog[w][(r + hi * 8) * 49 + col0 + hl] = acc[r];
    }
    __syncthreads();

    if (lane < 16) {                           // lane L: softmax of node n0+L
        int nd = n0 + lane;
        if (nd < N) {
            float m = -1e30f;
            for (int c = 0; c < K1; c++) {
                float v = slog[w][lane * 49 + c] + pb1[c];
                slog[w][lane * 49 + c] = v;
                m = fmaxf(m, v);
            }
            float sum = 0.0f;
            for (int c = 0; c < K1; c++) {
                float e = expf(slog[w][lane * 49 + c] - m);
                slog[w][lane * 49 + c] = e;
                sum += e;
            }
            float inv = 1.0f / sum;
            for (int c = 0; c < K1; c++)
                s[nd * K1 + c] = slog[w][lane * 49 + c] * inv;
        }
    }
}

// ---------------------------------------------------------------------------
// 5) out1[k,f] = sum_n s[n,k]*h1[n,f]   (LDS partial sums + global atomics)
// ---------------------------------------------------------------------------
__global__ void k_stx(const float* __restrict__ s, const float* __restrict__ h1,
                      float* __restrict__ out1, int N) {
    __shared__ float sv[8 * K1];
    __shared__ float hv[8 * H_DIM];
    __shared__ float acc[K1 * H_DIM];          // 1120
    int t = threadIdx.x;
    for (int c = t; c < K1 * H_DIM; c += 256) acc[c] = 0.0f;
    int chunk = (N + gridDim.x - 1) / gridDim.x;
    int nA = blockIdx.x * chunk;
    int nB = min(nA + chunk, N);
    __syncthreads();
    for (int nb = nA; nb < nB; nb += 8) {
        for (int i = t; i < 8 * K1; i += 256) {
            int n = nb + i / K1;
            sv[i] = (n < nB) ? s[n * K1 + i % K1] : 0.0f;
        }
        for (int i = t; i < 8 * H_DIM; i += 256) {
            int n = nb + (i >> 5);
            hv[i] = (n < nB) ? h1[n * H_DIM + (i & 31)] : 0.0f;
        }
        __syncthreads();
        for (int c = t; c < K1 * H_DIM; c += 256) {
            int k = c >> 5, f = c & 31;
            float sum = 0.0f;
            #pragma unroll
            for (int j = 0; j < 8; j++) sum += sv[j * K1 + k] * hv[j * H_DIM + f];
            acc[c] += sum;
        }
        __syncthreads();
    }
    for (int c = t; c < K1 * H_DIM; c += 256) atomicAdd(&out1[c], acc[c]);
}

// ---------------------------------------------------------------------------
// 6) a_s[r,l] += s[c,l] per (edge,l)  ==  (sparse adj with multiplicity) @ S
// ---------------------------------------------------------------------------
__global__ void k_as(const int* __restrict__ e0, const int* __restrict__ e1,
                     const float* __restrict__ s, float* __restrict__ a_s, int E) {
    unsigned idx = blockIdx.x * 256u + threadIdx.x;
    unsigned e = idx / K1;
    unsigned l = idx % K1;
    if (e >= (unsigned)E) return;
    int r = e0[e], c = e1[e];
    atomicAdd(&a_s[r * K1 + l], s[c * K1 + l]);
}

// ---------------------------------------------------------------------------
// 7) adj1[k,l] = sum_n s[n,k]*a_s[n,l]
// ---------------------------------------------------------------------------
__global__ void k_sta(const float* __restrict__ s, const float* __restrict__ a_s,
                      float* __restrict__ adj1, int N) {
    __shared__ float sv[8 * K1];
    __shared__ float av[8 * K1];
    __shared__ float acc[K1 * K1];             // 1225
    int t = threadIdx.x;
    for (int c = t; c < K1 * K1; c += 256) acc[c] = 0.0f;
    int chunk = (N + gridDim.x - 1) / gridDim.x;
    int nA = blockIdx.x * chunk;
    int nB = min(nA + chunk, N);
    __syncthreads();
    for (int nb = nA; nb < nB; nb += 8) {
        for (int i = t; i < 8 * K1; i += 256) {
            int n = nb + i / K1;
            int q = i % K1;
            sv[i] = (n < nB) ? s[n * K1 + q]   : 0.0f;
            av[i] = (n < nB) ? a_s[n * K1 + q] : 0.0f;
        }
        __syncthreads();
        for (int c = t; c < K1 * K1; c += 256) {
            int k = c / K1, l = c % K1;
            float sum = 0.0f;
            #pragma unroll
            for (int j = 0; j < 8; j++) sum += sv[j * K1 + k] * av[j * K1 + l];
            acc[c] += sum;
        }
        __syncthreads();
    }
    for (int c = t; c < K1 * K1; c += 256) atomicAdd(&adj1[c], acc[c]);
}

// ---------------------------------------------------------------------------
// 8) tail: everything on the pooled (35 / 18 node) graphs, one workgroup.
// ---------------------------------------------------------------------------
__global__ void k_tail(const float* __restrict__ adj1g, const float* __restrict__ out1g,
                       const float* __restrict__ W2,  const float* __restrict__ b2,
                       const float* __restrict__ pW2, const float* __restrict__ pb2,
                       const float* __restrict__ W3,  const float* __restrict__ b3,
                       const float* __restrict__ l1W, const float* __restrict__ l1b,
                       const float* __restrict__ l2W, const float* __restrict__ l2b,
                       float* __restrict__ out) {
    __shared__ float A[K1 * K1], An[K1 * K1];
    __shared__ float X[K1 * H_DIM], T[K1 * H_DIM], X2[K1 * H_DIM];
    __shared__ float S2[K1 * K2], AS2[K1 * K2];
    __shared__ float A2[K2 * K2], A2n[K2 * K2];
    __shared__ float O2[K2 * H_DIM], T2[K2 * H_DIM], X3[K2 * H_DIM];
    __shared__ float d1[K1], di1[K1], dd[K2], di3[K2], g[64], hm[32];
    int t = threadIdx.x;

    for (int c = t; c < K1 * K1; c += 256) A[c] = adj1g[c];
    for (int c = t; c < K1 * H_DIM; c += 256) X[c] = out1g[c];
    __syncthreads();
    // ---- mincut pool 1 normalize: diag<-0, d = sqrt(rowsum)+eps, A/(d_k d_l)
    if (t < K1) A[t * (K1 + 1)] = 0.0f;
    __syncthreads();
    if (t < K1) {
        float sm = 0.0f;
        for (int l = 0; l < K1; l++) sm += A[t * K1 + l];
        d1[t] = sqrtf(sm) + 1e-15f;
    }
    __syncthreads();
    for (int c = t; c < K1 * K1; c += 256) {
        int k = c / K1, l = c % K1;
        An[c] = A[c] / (d1[k] * d1[l]);
    }
    __syncthreads();
    // ---- dense GCN conv (W2): diag<-1, di = rsqrt(max(rowsum,1)), relu
    if (t < K1) {
        float sm = 1.0f;                        // diag of An is 0; conv sets it to 1
        for (int l = 0; l < K1; l++) sm += An[t * K1 + l];
        di1[t] = rsqrtf(fmaxf(sm, 1.0f));
    }
    for (int c = t; c < K1 * H_DIM; c += 256) { // T = X @ W2
        int i = c >> 5, j = c & 31;
        float sm = 0.0f;
        for (int k = 0; k < H_DIM; k++) sm += X[i * H_DIM + k] * W2[k * H_DIM + j];
        T[c] = sm;
    }
    __syncthreads();
    for (int c = t; c < K1 * H_DIM; c += 256) { // X2 = relu(Bn @ T + b2)
        int i = c >> 5, j = c & 31;
        float sm = 0.0f;
        for (int k = 0; k < K1; k++) {
            float a = An[i * K1 + k] + (k == i ? 1.0f : 0.0f);
            sm += di1[i] * di1[k] * a * T[k * H_DIM + j];
        }
        X2[c] = fmaxf(sm + b2[j], 0.0f);
    }
    __syncthreads();
    // ---- s2 = softmax(X2 @ pW2 + pb2)
    for (int c = t; c < K1 * K2; c += 256) {
        int i = c / K2, j = c % K2;
        float sm = 0.0f;
        for (int k = 0; k < H_DIM; k++) sm += X2[i * H_DIM + k] * pW2[k * K2 + j];
        S2[c] = sm + pb2[j];
    }
    __syncthreads();
    if (t < K1) {
        float m = -1e30f;
        for (int j = 0; j < K2; j++) m = fmaxf(m, S2[t * K2 + j]);
        float sm = 0.0f;
        for (int j = 0; j < K2; j++) { float e = expf(S2[t * K2 + j] - m); S2[t * K2 + j] = e; sm += e; }
        float inv = 1.0f / sm;
        for (int j = 0; j < K2; j++) S2[t * K2 + j] *= inv;
    }
    __syncthreads();
    // ---- mincut pool 2: O2 = S2^T X2 ; AS2 = An @ S2 ; A2 = S2^T AS2
    for (int c = t; c < K2 * H_DIM; c += 256) {
        int k = c >> 5, f = c & 31;
        float sm = 0.0f;
        for (int n = 0; n < K1; n++) sm += S2[n * K2 + k] * X2[n * H_DIM + f];
        O2[c] = sm;
    }
    for (int c = t; c < K1 * K2; c += 256) {
        int i = c / K2, j = c % K2;
        float sm = 0.0f;
        for (int m = 0; m < K1; m++) sm += An[i * K1 + m] * S2[m * K2 + j];
        AS2[c] = sm;
    }
    __syncthreads();
    for (int c = t; c < K2 * K2; c += 256) {
        int k = c / K2, l = c % K2;
        float sm = 0.0f;
        for (int n = 0; n < K1; n++) sm += S2[n * K2 + k] * AS2[n * K2 + l];
        A2[c] = sm;
    }
    __syncthreads();
    if (t < K2) A2[t * (K2 + 1)] = 0.0f;
    __syncthreads();
    if (t < K2) {
        float sm = 0.0f;
        for (int l = 0; l < K2; l++) sm += A2[t * K2 + l];
        dd[t] = sqrtf(sm) + 1e-15f;
    }
    __syncthreads();
    for (int c = t; c < K2 * K2; c += 256) {
        int k = c / K2, l = c % K2;
        A2n[c] = A2[c] / (dd[k] * dd[l]);
    }
    __syncthreads();
    // ---- dense GCN conv (W3), no relu
    if (t < K2) {
        float sm = 1.0f;
        for (int l = 0; l < K2; l++) sm += A2n[t * K2 + l];
        di3[t] = rsqrtf(fmaxf(sm, 1.0f));
    }
    for (int c = t; c < K2 * H_DIM; c += 256) {
        int i = c >> 5, j = c & 31;
        float sm = 0.0f;
        for (int k = 0; k < H_DIM; k++) sm += O2[i * H_DIM + k] * W3[k * H_DIM + j];
        T2[c] = sm;
    }
    __syncthreads();
    for (int c = t; c < K2 * H_DIM; c += 256) {
        int i = c >> 5, j = c & 31;
        float sm = 0.0f;
        for (int k = 0; k < K2; k++) {
            float a = A2n[i * K2 + k] + (k == i ? 1.0f : 0.0f);
            sm += di3[i] * di3[k] * a * T2[k * H_DIM + j];
        }
        X3[c] = sm + b3[j];
    }
    __syncthreads();
    // ---- readout: g = [mean, sum]; out = relu(g@l1W+l1b)@l2W+l2b
    if (t < 32) {
        float sm = 0.0f;
        for (int n = 0; n < K2; n++) sm += X3[n * H_DIM + t];
        g[t] = sm / (float)K2;
        g[32 + t] = sm;
    }
    __syncthreads();
    if (t < 32) {
        float sm = 0.0f;
        for (int k = 0; k < 64; k++) sm += g[k] * l1W[k * 32 + t];
        hm[t] = fmaxf(sm + l1b[t], 0.0f);
    }
    __syncthreads();
    if (t < 2) {
        float sm = 0.0f;
        for (int k = 0; k < 32; k++) sm += hm[k] * l2W[k * 2 + t];
        out[t] = sm + l2b[t];
    }
}

// ---------------------------------------------------------------------------
extern "C" void kernel_launch(void* const* d_in, const int* in_sizes, int n_in,
                              void* d_out, int out_size, void* d_ws, size_t ws_size,
                              hipStream_t stream) {
    const float* x   = (const float*)d_in[0];
    const int*   ei  = (const int*)  d_in[1];
    const float* W1  = (const float*)d_in[2];
    const float* b1  = (const float*)d_in[3];
    const float* pW1 = (const float*)d_in[4];
    const float* pb1 = (const float*)d_in[5];
    const float* W2  = (const float*)d_in[6];
    const float* b2  = (const float*)d_in[7];
    const float* pW2 = (const float*)d_in[8];
    const float* pb2 = (const float*)d_in[9];
    const float* W3  = (const float*)d_in[10];
    const float* b3  = (const float*)d_in[11];
    const float* l1W = (const float*)d_in[12];
    const float* l1b = (const float*)d_in[13];
    const float* l2W = (const float*)d_in[14];
    const float* l2b = (const float*)d_in[15];

    const int N = in_sizes[0] / F_IN;
    const int E = in_sizes[1] / 2;
    const int* e0 = ei;
    const int* e1 = ei + E;

    // workspace layout (floats)
    float* ws   = (float*)d_ws;
    float* deg  = ws;                       // N
    float* dis  = deg  + N;                 // N
    float* h    = dis  + N;                 // N*32
    float* agg  = h    + (size_t)N * H_DIM; // N*32
    float* h1   = agg  + (size_t)N * H_DIM; // N*32
    float* s    = h1   + (size_t)N * H_DIM; // N*35
    float* a_s  = s    + (size_t)N * K1;    // N*35
    float* out1 = a_s  + (size_t)N * K1;    // 35*32
    float* adj1 = out1 + K1 * H_DIM;        // 35*35
    size_t need = ((size_t)(adj1 - ws) + K1 * K1) * sizeof(float);
    if (ws_size < need) return;             // safety valve: cannot run correctly

    float* outp = (float*)d_out;

    const int initTot = N + N * K1 + K1 * H_DIM + K1 * K1;
    const int nt64 = (N + 63) / 64;         // 4 wave-tiles (64 nodes) per block
    k_init<<<(initTot + 255) / 256, 256, 0, stream>>>(deg, a_s, out1, adj1, N);
    k_deg <<<(E + 255) / 256, 256, 0, stream>>>(e1, deg, E);
    k_dis <<<(N + 255) / 256, 256, 0, stream>>>(deg, dis, N);
    k_xw1 <<<nt64, 128, 0, stream>>>(x, W1, dis, h, agg, N);
    {
        long tot = (long)E * H_DIM;
        k_edge<<<(unsigned)((tot + 255) / 256), 256, 0, stream>>>(e0, e1, dis, h, agg, E);
    }
    k_h1s <<<nt64, 128, 0, stream>>>(agg, b1, pW1, pb1, h1, s, N);
    k_stx <<<120, 256, 0, stream>>>(s, h1, out1, N);
    {
        long tot = (long)E * K1;
        k_as <<<(unsigned)((tot + 255) / 256), 256, 0, stream>>>(e0, e1, s, a_s, E);
    }
    k_sta <<<120, 256, 0, stream>>>(s, a_s, adj1, N);
    k_tail<<<1, 256, 0, stream>>>(adj1, out1, W2, b2, pW2, pb2, W3, b3,
                                  l1W, l1b, l2W, l2b, outp);
}